// GATNetwork_56049323213493
// MI455X (gfx1250) — compile-verified
//
#include <hip/hip_runtime.h>
#include <math.h>

#define NNODES 100000
#define NEDGES 1600000
#define ETOT   (NEDGES + NNODES)   // with self-loops
#define DIN    256
#define HH     4
#define CC     16
#define HC     64                  // H*C
#define DOUT   40
#define NEG_SLOPE 0.2f

typedef __attribute__((ext_vector_type(2))) float v2f;
typedef __attribute__((ext_vector_type(8))) float v8f;

__device__ __forceinline__ float leaky(float x) { return x > 0.f ? x : NEG_SLOPE * x; }

// float atomic max via sign-split int/uint atomics (M initialized to -inf)
__device__ __forceinline__ void atomicMaxF(float* addr, float val) {
  if (val >= 0.f) atomicMax((int*)addr, __float_as_int(val));
  else            atomicMin((unsigned int*)addr, __float_as_uint(val));
}

__device__ __forceinline__ void edge_sd(size_t e, const long long* __restrict__ ei,
                                        int& s, int& d) {
  if (e < (size_t)NEDGES) { s = (int)ei[e]; d = (int)ei[(size_t)NEDGES + e]; }
  else                    { s = d = (int)(e - (size_t)NEDGES); }
}

// ---------------------------------------------------------------- utility
__global__ void fill_kernel(float* __restrict__ p, float v, size_t n) {
  size_t i = (size_t)blockIdx.x * blockDim.x + threadIdx.x;
  if (i < n) p[i] = v;
}

// ------------------------------------------------- GEMM1: XP1 = X @ W1  (fp32 WMMA)
// one wave per 16x16 output tile; K=256 in steps of 4 (V_WMMA_F32_16X16X4_F32)
__global__ void gemm_x_w1(const float* __restrict__ X, const float* __restrict__ W,
                          float* __restrict__ XP) {
  const int lane  = threadIdx.x & 31;
  const int wid   = blockIdx.x * (blockDim.x >> 5) + (threadIdx.x >> 5);
  const int ntile = wid & 3;          // 4 tiles of 16 cols (== one head each)
  const int rb    = wid >> 2;         // row block (N/16 = 6250 exact)
  if (rb >= NNODES / 16) return;
  const int half = lane >> 4;         // lanes 16-31 hold K+2/K+3 (A) and rows M+8 (D)
  const int l    = lane & 15;
  const float* xrow = X + (size_t)(rb * 16 + l) * DIN;
  const int ncol = ntile * 16 + l;
  v8f acc = {};
  for (int k = 0; k < DIN; k += 4) {
    float4 xa = *(const float4*)(xrow + k);
    v2f a, b;
    a.x = half ? xa.z : xa.x;         // A[M=l, K=k+2*half]
    a.y = half ? xa.w : xa.y;         // A[M=l, K=k+2*half+1]
    const int kb = k + (half << 1);
    b.x = W[(size_t)kb * HC + ncol];          // B[K, N=ncol]
    b.y = W[(size_t)(kb + 1) * HC + ncol];
    acc = __builtin_amdgcn_wmma_f32_16x16x4_f32(false, a, false, b,
                                                (short)0, acc, false, false);
  }
  // D layout: VGPR j -> row rb*16 + half*8 + j, col ncol
  float* op = XP + (size_t)(rb * 16 + half * 8) * HC + ncol;
#pragma unroll
  for (int j = 0; j < 8; ++j) op[(size_t)j * HC] = acc[j];
}

// ------------------------------------------------- GEMM2: XP2 = H @ W2 (64 -> 40)
__global__ void gemm_h_w2(const float* __restrict__ Hm, const float* __restrict__ W,
                          float* __restrict__ XP) {
  const int lane  = threadIdx.x & 31;
  const int wid   = blockIdx.x * (blockDim.x >> 5) + (threadIdx.x >> 5);
  const int ntile = wid % 3;          // cols 0-15, 16-31, 32-47(masked to <40)
  const int rb    = wid / 3;
  if (rb >= NNODES / 16) return;
  const int half = lane >> 4;
  const int l    = lane & 15;
  const float* hrow = Hm + (size_t)(rb * 16 + l) * HC;
  const int ncol = ntile * 16 + l;
  const bool valid = ncol < DOUT;
  v8f acc = {};
  for (int k = 0; k < HC; k += 4) {
    float4 ha = *(const float4*)(hrow + k);
    v2f a, b;
    a.x = half ? ha.z : ha.x;
    a.y = half ? ha.w : ha.y;
    const int kb = k + (half << 1);
    b.x = valid ? W[(size_t)kb * DOUT + ncol] : 0.f;
    b.y = valid ? W[(size_t)(kb + 1) * DOUT + ncol] : 0.f;
    acc = __builtin_amdgcn_wmma_f32_16x16x4_f32(false, a, false, b,
                                                (short)0, acc, false, false);
  }
  if (valid) {
    float* op = XP + (size_t)(rb * 16 + half * 8) * DOUT + ncol;
#pragma unroll
    for (int j = 0; j < 8; ++j) op[(size_t)j * DOUT] = acc[j];
  }
}

// ---------------------------------------------- per-node attention coefficients
__global__ void attn_coef1(const float* __restrict__ XP, const float* __restrict__ as,
                           const float* __restrict__ ad, float* __restrict__ ES,
                           float* __restrict__ ED) {
  int i = blockIdx.x * blockDim.x + threadIdx.x;     // i = n*4 + h
  if (i >= NNODES * HH) return;
  int h = i & 3, n = i >> 2;
  const float* p = XP + (size_t)n * HC + h * CC;
  float es = 0.f, ed = 0.f;
#pragma unroll
  for (int c = 0; c < CC; ++c) {
    es += p[c] * as[h * CC + c];
    ed += p[c] * ad[h * CC + c];
  }
  ES[i] = es; ED[i] = ed;
}

__global__ void attn_coef2(const float* __restrict__ XP, const float* __restrict__ as,
                           const float* __restrict__ ad, float* __restrict__ ES,
                           float* __restrict__ ED) {
  int n = blockIdx.x * blockDim.x + threadIdx.x;
  if (n >= NNODES) return;
  const float* p = XP + (size_t)n * DOUT;
  float es = 0.f, ed = 0.f;
#pragma unroll
  for (int c = 0; c < DOUT; ++c) { es += p[c] * as[c]; ed += p[c] * ad[c]; }
  ES[n] = es; ED[n] = ed;
}

// ---------------------------------------------- edge passes, layer 1 (H=4)
__global__ void edge_max1(const long long* __restrict__ ei, const float* __restrict__ ES,
                          const float* __restrict__ ED, float* __restrict__ M) {
  size_t i = (size_t)blockIdx.x * blockDim.x + threadIdx.x;
  if (i >= (size_t)ETOT * HH) return;
  int h = (int)(i & 3); size_t e = i >> 2;
  int s, d; edge_sd(e, ei, s, d);
  float lr = leaky(ES[(size_t)s * HH + h] + ED[(size_t)d * HH + h]);
  atomicMaxF(&M[(size_t)d * HH + h], lr);
}

__global__ void edge_sum1(const long long* __restrict__ ei, const float* __restrict__ ES,
                          const float* __restrict__ ED, const float* __restrict__ M,
                          float* __restrict__ Z) {
  size_t i = (size_t)blockIdx.x * blockDim.x + threadIdx.x;
  if (i >= (size_t)ETOT * HH) return;
  int h = (int)(i & 3); size_t e = i >> 2;
  int s, d; edge_sd(e, ei, s, d);
  float lr = leaky(ES[(size_t)s * HH + h] + ED[(size_t)d * HH + h]);
  float m = M[(size_t)d * HH + h]; m = (m > -3.0e38f) ? m : 0.f;   // isfinite guard
  atomicAdd(&Z[(size_t)d * HH + h], __expf(lr - m));
}

__global__ void edge_agg1(const long long* __restrict__ ei, const float* __restrict__ ES,
                          const float* __restrict__ ED, const float* __restrict__ M,
                          const float* __restrict__ Z, const float* __restrict__ XP,
                          float* __restrict__ O) {
  size_t i = (size_t)blockIdx.x * blockDim.x + threadIdx.x;
  if (i >= (size_t)ETOT * HC) return;
  int c = (int)(i & 63); size_t e = i >> 6;
  int h = c >> 4;
  int s, d; edge_sd(e, ei, s, d);
  float lr = leaky(ES[(size_t)s * HH + h] + ED[(size_t)d * HH + h]);
  float m = M[(size_t)d * HH + h]; m = (m > -3.0e38f) ? m : 0.f;
  float alpha = __expf(lr - m) / (Z[(size_t)d * HH + h] + 1e-16f);
  atomicAdd(&O[(size_t)d * HC + c], XP[(size_t)s * HC + c] * alpha);
}

// ---------------------------------------------- edge passes, layer 2 (H=1)
__global__ void edge_max2(const long long* __restrict__ ei, const float* __restrict__ ES,
                          const float* __restrict__ ED, float* __restrict__ M) {
  size_t e = (size_t)blockIdx.x * blockDim.x + threadIdx.x;
  if (e >= (size_t)ETOT) return;
  int s, d; edge_sd(e, ei, s, d);
  atomicMaxF(&M[d], leaky(ES[s] + ED[d]));
}

__global__ void edge_sum2(const long long* __restrict__ ei, const float* __restrict__ ES,
                          const float* __restrict__ ED, const float* __restrict__ M,
                          float* __restrict__ Z) {
  size_t e = (size_t)blockIdx.x * blockDim.x + threadIdx.x;
  if (e >= (size_t)ETOT) return;
  int s, d; edge_sd(e, ei, s, d);
  float lr = leaky(ES[s] + ED[d]);
  float m = M[d]; m = (m > -3.0e38f) ? m : 0.f;
  atomicAdd(&Z[d], __expf(lr - m));
}

__global__ void edge_agg2(const long long* __restrict__ ei, const float* __restrict__ ES,
                          const float* __restrict__ ED, const float* __restrict__ M,
                          const float* __restrict__ Z, const float* __restrict__ XP,
                          float* __restrict__ O) {
  size_t i = (size_t)blockIdx.x * blockDim.x + threadIdx.x;
  if (i >= (size_t)ETOT * DOUT) return;
  int c = (int)(i % DOUT); size_t e = i / DOUT;
  int s, d; edge_sd(e, ei, s, d);
  float lr = leaky(ES[s] + ED[d]);
  float m = M[d]; m = (m > -3.0e38f) ? m : 0.f;
  float alpha = __expf(lr - m) / (Z[d] + 1e-16f);
  atomicAdd(&O[(size_t)d * DOUT + c], XP[(size_t)s * DOUT + c] * alpha);
}

// ---------------------------------------------- epilogues
__global__ void relu_bias1(float* __restrict__ O, const float* __restrict__ b) {
  size_t i = (size_t)blockIdx.x * blockDim.x + threadIdx.x;
  if (i >= (size_t)NNODES * HC) return;
  float v = O[i] + b[i & 63];
  O[i] = v > 0.f ? v : 0.f;
}

__global__ void add_bias2(float* __restrict__ O, const float* __restrict__ b) {
  size_t i = (size_t)blockIdx.x * blockDim.x + threadIdx.x;
  if (i >= (size_t)NNODES * DOUT) return;
  O[i] += b[i % DOUT];
}

static inline unsigned nblk(size_t n) { return (unsigned)((n + 255) / 256); }

extern "C" void kernel_launch(void* const* d_in, const int* in_sizes, int n_in,
                              void* d_out, int out_size, void* d_ws, size_t ws_size,
                              hipStream_t stream) {
  const float*     x   = (const float*)d_in[0];
  const long long* ei  = (const long long*)d_in[1];   // int64 edge_index [2,E]
  const float*     W1  = (const float*)d_in[2];
  const float*     as1 = (const float*)d_in[3];
  const float*     ad1 = (const float*)d_in[4];
  const float*     b1  = (const float*)d_in[5];
  const float*     W2  = (const float*)d_in[6];
  const float*     as2 = (const float*)d_in[7];
  const float*     ad2 = (const float*)d_in[8];
  const float*     b2  = (const float*)d_in[9];
  float* out = (float*)d_out;

  // workspace layout (floats): [XP1 (N*64) | ES1,ED1,M1,Z1 (N*4 each) | O1/H1 (N*64)]
  // XP2(N*40)+ES2/ED2/M2/Z2(N each) alias the XP1 region once layer 1 is done.
  float* ws  = (float*)d_ws;
  float* XP1 = ws;
  float* XP2 = ws;
  float* ES2 = ws + (size_t)NNODES * DOUT;
  float* ED2 = ES2 + NNODES;
  float* M2  = ED2 + NNODES;
  float* Z2  = M2 + NNODES;
  float* ES1 = ws + (size_t)NNODES * HC;
  float* ED1 = ES1 + (size_t)NNODES * HH;
  float* M1  = ED1 + (size_t)NNODES * HH;
  float* Z1  = M1 + (size_t)NNODES * HH;
  float* O1  = Z1 + (size_t)NNODES * HH;

  const size_t NH = (size_t)NNODES * HH, NHC = (size_t)NNODES * HC;
  const size_t NO = (size_t)NNODES * DOUT;
  const size_t E4 = (size_t)ETOT * HH, E64 = (size_t)ETOT * HC, E40 = (size_t)ETOT * DOUT;

  // ---------------- layer 1 ----------------
  fill_kernel<<<nblk(NH), 256, 0, stream>>>(M1, -INFINITY, NH);
  fill_kernel<<<nblk(NH), 256, 0, stream>>>(Z1, 0.f, NH);
  fill_kernel<<<nblk(NHC), 256, 0, stream>>>(O1, 0.f, NHC);
  gemm_x_w1<<<(NNODES / 16) * 4 / 8, 256, 0, stream>>>(x, W1, XP1);   // 8 waves/block
  attn_coef1<<<nblk(NH), 256, 0, stream>>>(XP1, as1, ad1, ES1, ED1);
  edge_max1<<<nblk(E4), 256, 0, stream>>>(ei, ES1, ED1, M1);
  edge_sum1<<<nblk(E4), 256, 0, stream>>>(ei, ES1, ED1, M1, Z1);
  edge_agg1<<<nblk(E64), 256, 0, stream>>>(ei, ES1, ED1, M1, Z1, XP1, O1);
  relu_bias1<<<nblk(NHC), 256, 0, stream>>>(O1, b1);

  // ---------------- layer 2 ----------------
  gemm_h_w2<<<((NNODES / 16) * 3 + 7) / 8, 256, 0, stream>>>(O1, W2, XP2);
  attn_coef2<<<nblk(NNODES), 256, 0, stream>>>(XP2, as2, ad2, ES2, ED2);
  fill_kernel<<<nblk(NNODES), 256, 0, stream>>>(M2, -INFINITY, NNODES);
  fill_kernel<<<nblk(NNODES), 256, 0, stream>>>(Z2, 0.f, NNODES);
  fill_kernel<<<nblk(NO), 256, 0, stream>>>(out, 0.f, NO);
  edge_max2<<<nblk(ETOT), 256, 0, stream>>>(ei, ES2, ED2, M2);
  edge_sum2<<<nblk(ETOT), 256, 0, stream>>>(ei, ES2, ED2, M2, Z2);
  edge_agg2<<<nblk(E40), 256, 0, stream>>>(ei, ES2, ED2, M2, Z2, XP2, out);
  add_bias2<<<nblk(NO), 256, 0, stream>>>(out, b2);
}